// GraphTransformerLayer_40587440947793
// MI455X (gfx1250) — compile-verified
//
#include <hip/hip_runtime.h>
#include <math.h>

// ---------------- CDNA5 WMMA (fp32, 16x16x4) ----------------
typedef float v2f __attribute__((ext_vector_type(2)));
typedef float v8f __attribute__((ext_vector_type(8)));

#define NN   2048
#define DD   128
#define HH   8
#define DHH  16
#define EE   65536
#define FFD  512
#define EPSL 1e-5f
#define AW   (NN/32)   // adjacency words per row

__device__ __forceinline__ v8f wmma4(v2f a, v2f b, v8f c) {
  // D = A(16x4,f32) * B(4x16,f32) + C(16x16,f32)
  return __builtin_amdgcn_wmma_f32_16x16x4_f32(false, a, false, b, (short)0, c,
                                               false, false);
}

// ---------------- Generic fp32 WMMA GEMM: C = A[MxK]@B[KxN] + bias (+resid) (relu?) ----
__global__ void wmma_gemm(const float* __restrict__ A, const float* __restrict__ B,
                          const float* __restrict__ bias, const float* __restrict__ resid,
                          float* __restrict__ C, int M, int Kd, int Nn, int relu) {
  const int lane = threadIdx.x;
  const int half = lane >> 4, l16 = lane & 15;
  const int j0 = blockIdx.x * 16, i0 = blockIdx.y * 16;
  const int m = i0 + l16;
  v8f acc = {0.f,0.f,0.f,0.f,0.f,0.f,0.f,0.f};
  for (int kc = 0; kc < Kd; kc += 4) {
    const int kb = kc + 2 * half;           // A-frag: lanes 0-15 hold k,k+1; 16-31 hold k+2,k+3
    v2f a; a[0] = A[m * Kd + kb];       a[1] = A[m * Kd + kb + 1];
    v2f b; b[0] = B[kb * Nn + j0 + l16]; b[1] = B[(kb + 1) * Nn + j0 + l16];
    acc = wmma4(a, b, acc);
  }
  const int j = j0 + l16;
  const float bj = bias[j];
  for (int r = 0; r < 8; ++r) {
    const int i = i0 + r + 8 * half;
    float v = acc[r] + bj;
    if (resid) v += resid[i * Nn + j];
    if (relu)  v = fmaxf(v, 0.f);
    C[i * Nn + j] = v;
  }
}

// ---------------- edge_bias = edge_feat[E,10] @ We[10,8] + be ----------------
__global__ void edge_bias_kernel(const float* __restrict__ EF, const float* __restrict__ We,
                                 const float* __restrict__ be, float* __restrict__ EB) {
  const int e = blockIdx.x * blockDim.x + threadIdx.x;
  if (e >= EE) return;
  float acc[HH];
  for (int h = 0; h < HH; ++h) acc[h] = be[h];
  for (int k = 0; k < 10; ++k) {
    const float f = EF[e * 10 + k];
    for (int h = 0; h < HH; ++h) acc[h] += f * We[k * HH + h];
  }
  for (int h = 0; h < HH; ++h) EB[e * HH + h] = acc[h];
}

// ---------------- zero adjacency bitmask + edge-id map ----------------
__global__ void zero_maps(unsigned* __restrict__ adj, int* __restrict__ eid) {
  const int t = blockIdx.x * blockDim.x + threadIdx.x;
  const int stride = gridDim.x * blockDim.x;
  for (int i = t; i < NN * NN; i += stride) eid[i] = 0;
  for (int i = t; i < NN * AW; i += stride) adj[i] = 0u;
}

// ---------------- scatter edges: adj bits (sym + diag), eid map (atomicMax = det.) ----
__global__ void scatter_maps(const int* __restrict__ src, const int* __restrict__ dst,
                             unsigned* __restrict__ adj, int* __restrict__ eid) {
  const int t = blockIdx.x * blockDim.x + threadIdx.x;
  if (t < EE) {
    const int s = src[t], d = dst[t];
    atomicMax(&eid[s * NN + d], t + 1);                 // last duplicate wins, deterministic
    atomicOr(&adj[s * AW + (d >> 5)], 1u << (d & 31));
    atomicOr(&adj[d * AW + (s >> 5)], 1u << (s & 31));
  }
  if (t < NN) atomicOr(&adj[t * AW + (t >> 5)], 1u << (t & 31)); // self loops
}

// ---------------- fused flash attention, one wave per (16-query tile, head) -----------
__global__ void attn_flash(const float* __restrict__ Q, const float* __restrict__ K,
                           const float* __restrict__ V, const float* __restrict__ EB,
                           const unsigned* __restrict__ adj, const int* __restrict__ eid,
                           float* __restrict__ O) {
  __shared__ float pt[256];                 // P tile staging (16x16)
  const int lane = threadIdx.x;
  const int half = lane >> 4, l16 = lane & 15;
  const int i0 = blockIdx.x * 16;
  const int h  = blockIdx.y;
  const int hd = h * DHH;

  // Q tile as A-fragments (K dim = DH = 16 -> 4 chunks of 4)
  v2f qa[4];
  for (int c = 0; c < 4; ++c) {
    const int kb = 4 * c + 2 * half;
    v2f t; t[0] = Q[(i0 + l16) * DD + hd + kb];
           t[1] = Q[(i0 + l16) * DD + hd + kb + 1];
    qa[c] = t;
  }

  float mrow[8], lsum[8];
  for (int r = 0; r < 8; ++r) { mrow[r] = -INFINITY; lsum[r] = 0.f; }
  v8f acc = {0.f,0.f,0.f,0.f,0.f,0.f,0.f,0.f};

  for (int jt = 0; jt < NN / 16; ++jt) {
    const int j0 = jt * 16;
    // S = Q * K^T  (B-frag: lane = key column j, VGPRs = d-chunks)
    v8f s = {0.f,0.f,0.f,0.f,0.f,0.f,0.f,0.f};
    for (int c = 0; c < 4; ++c) {
      const int kb = 4 * c + 2 * half;
      v2f kf; kf[0] = K[(j0 + l16) * DD + hd + kb];
              kf[1] = K[(j0 + l16) * DD + hd + kb + 1];
      s = wmma4(qa[c], kf, s);
    }
    const int j = j0 + l16;
    const int wcol = j >> 5;
    const unsigned jbit = 1u << (j & 31);

    // mask + edge bias + online softmax (row i lives across one 16-lane half)
    for (int r = 0; r < 8; ++r) {
      const int i = i0 + r + 8 * half;
      float sv = s[r] * 0.25f;                       // 1/sqrt(16)
      if (adj[i * AW + wcol] & jbit) {
        const int e = eid[i * NN + j];
        if (e > 0) sv += EB[(e - 1) * HH + h];
      } else {
        sv = -INFINITY;
      }
      float rm = sv;
      rm = fmaxf(rm, __shfl_xor(rm, 1, 32));
      rm = fmaxf(rm, __shfl_xor(rm, 2, 32));
      rm = fmaxf(rm, __shfl_xor(rm, 4, 32));
      rm = fmaxf(rm, __shfl_xor(rm, 8, 32));
      const float mn = fmaxf(mrow[r], rm);
      float alpha, p;
      if (mn == -INFINITY) { alpha = 1.f; p = 0.f; }
      else { alpha = __expf(mrow[r] - mn); p = __expf(sv - mn); }
      mrow[r] = mn;
      float ps = p;
      ps += __shfl_xor(ps, 1, 32);
      ps += __shfl_xor(ps, 2, 32);
      ps += __shfl_xor(ps, 4, 32);
      ps += __shfl_xor(ps, 8, 32);
      lsum[r] = lsum[r] * alpha + ps;
      acc[r] = acc[r] * alpha;
      pt[(r + 8 * half) * 16 + l16] = p;             // stage P for layout change
    }
    __syncthreads();
    // O += P * V   (P reloaded as A-fragments; V as B-fragments, lane = output dim d)
    for (int c = 0; c < 4; ++c) {
      const int kb = 4 * c + 2 * half;
      v2f pa; pa[0] = pt[l16 * 16 + kb];
              pa[1] = pt[l16 * 16 + kb + 1];
      v2f vb; vb[0] = V[(j0 + kb) * DD + hd + l16];
              vb[1] = V[(j0 + kb + 1) * DD + hd + l16];
      acc = wmma4(pa, vb, acc);
    }
    __syncthreads();
  }
  for (int r = 0; r < 8; ++r) {
    const int i = i0 + r + 8 * half;
    O[i * DD + hd + l16] = acc[r] / lsum[r];
  }
}

// ---------------- row LayerNorm over 128 cols, one wave per row ----------------
__global__ void ln_rows(const float* __restrict__ X, const float* __restrict__ g,
                        const float* __restrict__ b, float* __restrict__ Y) {
  const int row = blockIdx.x;
  const int lane = threadIdx.x;
  float x[4];
  float s = 0.f;
  for (int k = 0; k < 4; ++k) { x[k] = X[row * DD + lane + 32 * k]; s += x[k]; }
  for (int d = 1; d < 32; d <<= 1) s += __shfl_xor(s, d, 32);
  const float mu = s * (1.f / DD);
  float vs = 0.f;
  for (int k = 0; k < 4; ++k) { const float t = x[k] - mu; vs += t * t; }
  for (int d = 1; d < 32; d <<= 1) vs += __shfl_xor(vs, d, 32);
  const float inv = rsqrtf(vs * (1.f / DD) + EPSL);
  for (int k = 0; k < 4; ++k) {
    const int c = lane + 32 * k;
    Y[row * DD + c] = (x[k] - mu) * inv * g[c] + b[c];
  }
}

// ---------------- launch ----------------
extern "C" void kernel_launch(void* const* d_in, const int* in_sizes, int n_in,
                              void* d_out, int out_size, void* d_ws, size_t ws_size,
                              hipStream_t stream) {
  (void)in_sizes; (void)n_in; (void)out_size; (void)ws_size;
  const float* node = (const float*)d_in[0];
  const float* ef   = (const float*)d_in[1];
  const int*   src  = (const int*)d_in[2];   // jax x64 disabled -> int32
  const int*   dst  = (const int*)d_in[3];
  const float* Wq = (const float*)d_in[4];  const float* bq  = (const float*)d_in[5];
  const float* Wk = (const float*)d_in[6];  const float* bk  = (const float*)d_in[7];
  const float* Wv = (const float*)d_in[8];  const float* bv  = (const float*)d_in[9];
  const float* Wo = (const float*)d_in[10]; const float* bo  = (const float*)d_in[11];
  const float* We = (const float*)d_in[12]; const float* be  = (const float*)d_in[13];
  const float* g1 = (const float*)d_in[14]; const float* b1n = (const float*)d_in[15];
  const float* g2 = (const float*)d_in[16]; const float* b2n = (const float*)d_in[17];
  const float* W1 = (const float*)d_in[18]; const float* bf1 = (const float*)d_in[19];
  const float* W2 = (const float*)d_in[20]; const float* bf2 = (const float*)d_in[21];
  float* out = (float*)d_out;

  char* w = (char*)d_ws;
  size_t o = 0;
  float*    Q   = (float*)(w + o); o += (size_t)NN * DD * 4;      // 1 MB
  float*    Km  = (float*)(w + o); o += (size_t)NN * DD * 4;      // 1 MB
  float*    V   = (float*)(w + o); o += (size_t)NN * DD * 4;      // 1 MB
  float*    S   = (float*)(w + o); o += (size_t)NN * DD * 4;      // attn out, 1 MB
  float*    Y   = (float*)(w + o); o += (size_t)NN * DD * 4;      // pre-LN buf, 1 MB
  float*    X1  = (float*)(w + o); o += (size_t)NN * DD * 4;      // post-LN1, 1 MB
  float*    H1  = (float*)(w + o); o += (size_t)NN * FFD * 4;     // 4 MB
  float*    EB  = (float*)(w + o); o += (size_t)EE * HH * 4;      // 2 MB
  unsigned* ADJ = (unsigned*)(w + o); o += (size_t)NN * AW * 4;   // 512 KB
  int*      EID = (int*)(w + o); o += (size_t)NN * NN * 4;        // 16 MB

  const dim3 blk(32);
  // QKV projections (fp32 WMMA)
  wmma_gemm<<<dim3(DD/16, NN/16), blk, 0, stream>>>(node, Wq, bq, nullptr, Q,  NN, DD, DD, 0);
  wmma_gemm<<<dim3(DD/16, NN/16), blk, 0, stream>>>(node, Wk, bk, nullptr, Km, NN, DD, DD, 0);
  wmma_gemm<<<dim3(DD/16, NN/16), blk, 0, stream>>>(node, Wv, bv, nullptr, V,  NN, DD, DD, 0);
  // Edge bias table + sparse structure
  edge_bias_kernel<<<EE/256, 256, 0, stream>>>(ef, We, be, EB);
  zero_maps<<<1024, 256, 0, stream>>>(ADJ, EID);
  scatter_maps<<<EE/256, 256, 0, stream>>>(src, dst, ADJ, EID);
  // Fused masked-bias flash attention
  attn_flash<<<dim3(NN/16, HH), blk, 0, stream>>>(Q, Km, V, EB, ADJ, EID, S);
  // Output projection + residual, LN1
  wmma_gemm<<<dim3(DD/16, NN/16), blk, 0, stream>>>(S, Wo, bo, node, Y, NN, DD, DD, 0);
  ln_rows<<<NN, blk, 0, stream>>>(Y, g1, b1n, X1);
  // FFN
  wmma_gemm<<<dim3(FFD/16, NN/16), blk, 0, stream>>>(X1, W1, bf1, nullptr, H1, NN, DD, FFD, 1);
  wmma_gemm<<<dim3(DD/16, NN/16), blk, 0, stream>>>(H1, W2, bf2, X1, Y, NN, FFD, DD, 0);
  ln_rows<<<NN, blk, 0, stream>>>(Y, g2, b2n, out);
}